// Model_84146999263763
// MI455X (gfx1250) — compile-verified
//
#include <hip/hip_runtime.h>

// ---------------------------------------------------------------------------
// Problem constants (from reference)
// ---------------------------------------------------------------------------
#define BATCH   16
#define NPTS    16384
#define REGIONS 50
#define APAD    64          // anchors padded to 4 WMMA tiles of 16
#define TOPK    4
#define NEIGH   128
#define COURSE  512
#define R2      0.16f       // RADIUS^2
#define BALLK   1000
#define OUT_S   1536        // 512 roi1 + 512 roi2 + 512 fps

typedef float v2f __attribute__((ext_vector_type(2)));
typedef float v8f __attribute__((ext_vector_type(8)));

// ---------------------------------------------------------------------------
// Wave32 shuffle reductions (no barriers, no LDS round-trips).
// Tie-break: lowest global index, matching jnp.argmax / top_k semantics.
// ---------------------------------------------------------------------------
__device__ __forceinline__ void argmax_wave(float& v, int& i) {
#pragma unroll
    for (int m = 16; m > 0; m >>= 1) {
        float ov = __shfl_xor(v, m, 32);
        int   oi = __shfl_xor(i, m, 32);
        if (ov > v || (ov == v && oi < i)) { v = ov; i = oi; }
    }
}
__device__ __forceinline__ void argmin_wave(float& v, int& i) {
#pragma unroll
    for (int m = 16; m > 0; m >>= 1) {
        float ov = __shfl_xor(v, m, 32);
        int   oi = __shfl_xor(i, m, 32);
        if (ov < v || (ov == v && oi < i)) { v = ov; i = oi; }
    }
}

// ---------------------------------------------------------------------------
// Kernel 1: furthest point sampling, 512 steps, one block per batch.
// Points + running min-distance live in registers (16 pts/thread). Each step:
// register update -> wave32 shuffle argmax -> one LDS exchange of 32 wave
// partials reduced by wave 0 with shuffles. Only 2 block barriers per step
// (vs 10 for a full LDS tree) on the serial critical path.
// One FPS run feeds anchors (prefix of 50) and the final fps_pts (all 512).
// ---------------------------------------------------------------------------
__global__ __launch_bounds__(1024)
void fps_kernel(const float* __restrict__ pts, int* __restrict__ fps_idx) {
    const int b    = blockIdx.x;
    const int tid  = threadIdx.x;
    const int lane = tid & 31;
    const int wid  = tid >> 5;            // 32 waves
    const float* __restrict__ X = pts + (size_t)(b * 3 + 0) * NPTS;
    const float* __restrict__ Y = pts + (size_t)(b * 3 + 1) * NPTS;
    const float* __restrict__ Z = pts + (size_t)(b * 3 + 2) * NPTS;

    const int PPT = NPTS / 1024;          // 16 points per thread
    float lx[16], ly[16], lz[16], ld[16];
#pragma unroll
    for (int j = 0; j < PPT; ++j) {
        int i = tid + j * 1024;           // coalesced
        lx[j] = X[i]; ly[j] = Y[i]; lz[j] = Z[i];
        ld[j] = 1e10f;
    }

    __shared__ float sv[32];
    __shared__ int   si[32];
    __shared__ float cen[3];
    __shared__ int   win;
    if (tid == 0) win = 0;                // reference: first index is 0
    __syncthreads();

    for (int t = 0; t < COURSE; ++t) {
        if (tid == 0) {                   // emit index BEFORE the update
            int cur = win;
            fps_idx[b * COURSE + t] = cur;
            cen[0] = X[cur]; cen[1] = Y[cur]; cen[2] = Z[cur];
        }
        __syncthreads();                  // barrier 1: cen visible
        const float cx = cen[0], cy = cen[1], cz = cen[2];

        float best = -1.0f; int bi = 0;
#pragma unroll
        for (int j = 0; j < PPT; ++j) {
            float dx = lx[j] - cx, dy = ly[j] - cy, dz = lz[j] - cz;
            float d  = dx * dx + dy * dy + dz * dz;
            float nd = fminf(ld[j], d);
            ld[j] = nd;
            int i = tid + j * 1024;
            if (nd > best || (nd == best && i < bi)) { best = nd; bi = i; }
        }
        argmax_wave(best, bi);            // 5 shuffle steps, no barrier
        if (lane == 0) { sv[wid] = best; si[wid] = bi; }
        __syncthreads();                  // barrier 2: partials visible
        if (wid == 0) {                   // wave 0 reduces 32 partials
            float v = sv[lane]; int i2 = si[lane];
            argmax_wave(v, i2);
            if (lane == 0) win = i2;      // consumed next iter by tid 0 only
        }
        // no barrier needed: only tid 0 (in wave 0, after its own reduce)
        // reads `win`; other waves stall at barrier 1 before touching sv/si.
    }
}

// ---------------------------------------------------------------------------
// Kernel 2: gather anchor coords (first REGIONS FPS indices) + |a|^2, padded
// to APAD with a far sentinel so padded rows never count / never win top-4.
// ---------------------------------------------------------------------------
__global__ void anchor_kernel(const float* __restrict__ pts,
                              const int* __restrict__ fps_idx,
                              float* __restrict__ anch,
                              float* __restrict__ anorm) {
    const int b = blockIdx.x;
    const int a = threadIdx.x;            // 0..63
    const float* X = pts + (size_t)(b * 3 + 0) * NPTS;
    const float* Y = pts + (size_t)(b * 3 + 1) * NPTS;
    const float* Z = pts + (size_t)(b * 3 + 2) * NPTS;
    float x = 1e18f, y = 1e18f, z = 1e18f, nn = 1e37f;
    if (a < REGIONS) {
        int i = fps_idx[b * COURSE + a];
        x = X[i]; y = Y[i]; z = Z[i];
        nn = x * x + y * y + z * z;
    }
    float* row = anch + ((size_t)b * APAD + a) * 3;
    row[0] = x; row[1] = y; row[2] = z;
    anorm[(size_t)b * APAD + a] = nn;
}

// ---------------------------------------------------------------------------
// Kernel 3: ball counts via V_WMMA_F32_16X16X4_F32.
// One wave per (batch, 16-anchor tile). A-fragment (anchors, K=xyz0) stays
// resident; stream 16-point B-fragments over all 16384 points; epilogue fuses
// d^2 = |a|^2 + |p|^2 - 2*dot and the (< r^2) count so the full 50x16384
// distance matrix never leaves registers.
// A-frag layout (ISA 16x4 f32): lanes 0-15 -> K=0,1 ; lanes 16-31 -> K=2,3.
// C/D layout: element j of lane L is row M = j + 8*(L>>4), col N = L&15.
// ---------------------------------------------------------------------------
__global__ __launch_bounds__(32)
void count_kernel(const float* __restrict__ pts,
                  const float* __restrict__ anch,
                  const float* __restrict__ anorm,
                  int* __restrict__ counts) {
    const int b    = blockIdx.x;
    const int tile = blockIdx.y;          // 0..3
    const int lane = threadIdx.x;         // wave32
    const int hi   = lane >> 4;
    const int col  = lane & 15;

    const float* __restrict__ X = pts + (size_t)(b * 3 + 0) * NPTS;
    const float* __restrict__ Y = pts + (size_t)(b * 3 + 1) * NPTS;
    const float* __restrict__ Z = pts + (size_t)(b * 3 + 2) * NPTS;

    // A fragment: 16 anchors x (x,y,z,0)
    const float* arow = anch + ((size_t)b * APAD + tile * 16 + col) * 3;
    v2f afrag;
    afrag.x = (hi == 0) ? arow[0] : arow[2];   // K0=x / K2=z
    afrag.y = (hi == 0) ? arow[1] : 0.0f;      // K1=y / K3=0

    __shared__ float na_s[16];
    __shared__ int   cnt_s[16];
    if (lane < 16) {
        na_s[lane]  = anorm[(size_t)b * APAD + tile * 16 + lane];
        cnt_s[lane] = 0;
    }
    __syncthreads();

    int cnt[8];
#pragma unroll
    for (int j = 0; j < 8; ++j) cnt[j] = 0;

    for (int pt = 0; pt < NPTS / 16; ++pt) {
        const int n = pt * 16 + col;
        float px = X[n], py = Y[n], pz = Z[n];
        __builtin_prefetch(&X[n + 16], 0, 3);       // WGP-scope global_prefetch_b8
        float npn = px * px + py * py + pz * pz;

        v2f bfrag;                                  // 4x16 B tile, col = N
        bfrag.x = hi ? pz : px;
        bfrag.y = hi ? 0.0f : py;

        v8f c = {};
        v8f d = __builtin_amdgcn_wmma_f32_16x16x4_f32(
            /*neg_a=*/false, afrag, /*neg_b=*/false, bfrag,
            /*c_mod=*/(short)0, c, /*reuse_a=*/false, /*reuse_b=*/false);

#pragma unroll
        for (int j = 0; j < 8; ++j) {
            float d2 = na_s[j + 8 * hi] + npn - 2.0f * d[j];
            cnt[j] += (d2 < R2) ? 1 : 0;
        }
    }

#pragma unroll
    for (int j = 0; j < 8; ++j) atomicAdd(&cnt_s[j + 8 * hi], cnt[j]);
    __syncthreads();

    if (lane < 16) {
        int a = tile * 16 + lane;
        int v = cnt_s[lane];
        if (v > BALLK) v = BALLK;
        counts[(size_t)b * APAD + a] = (a < REGIONS) ? v : -1;
    }
}

// ---------------------------------------------------------------------------
// Kernel 4: top-4 anchor selection per batch (50 candidates; serial scan,
// ties -> lowest index, matching jax top_k).
// ---------------------------------------------------------------------------
__global__ void top4_kernel(const int* __restrict__ counts, int* __restrict__ sel) {
    int b = blockIdx.x * blockDim.x + threadIdx.x;
    if (b >= BATCH) return;
    int c[REGIONS];
    for (int i = 0; i < REGIONS; ++i) c[i] = counts[(size_t)b * APAD + i];
    for (int t = 0; t < TOPK; ++t) {
        int best = -0x7fffffff, bi = 0;
        for (int i = 0; i < REGIONS; ++i)
            if (c[i] > best) { best = c[i]; bi = i; }
        sel[b * TOPK + t] = bi;
        c[bi] = -0x7fffffff;
    }
}

// ---------------------------------------------------------------------------
// Kernel 5: 128-NN per (batch, selected anchor) by iterative min-extraction
// over an LDS taken-bitmask (matches top_k(-d) order: ascending distance,
// ties -> lowest index). Per pass: wave32 shuffle argmin + one LDS exchange
// of 8 wave partials (2 barriers/pass instead of 8). Writes the gathered
// points into BOTH roi1 and roi2 slots (reference computes regions twice).
// ---------------------------------------------------------------------------
__global__ __launch_bounds__(256)
void knn_kernel(const float* __restrict__ pts,
                const float* __restrict__ anch,
                const int* __restrict__ sel,
                float* __restrict__ out) {
    const int b    = blockIdx.x;
    const int t    = blockIdx.y;          // 0..3
    const int tid  = threadIdx.x;
    const int lane = tid & 31;
    const int wid  = tid >> 5;            // 8 waves
    const float* __restrict__ X = pts + (size_t)(b * 3 + 0) * NPTS;
    const float* __restrict__ Y = pts + (size_t)(b * 3 + 1) * NPTS;
    const float* __restrict__ Z = pts + (size_t)(b * 3 + 2) * NPTS;

    const int a = sel[b * TOPK + t];
    const float ax = anch[((size_t)b * APAD + a) * 3 + 0];
    const float ay = anch[((size_t)b * APAD + a) * 3 + 1];
    const float az = anch[((size_t)b * APAD + a) * 3 + 2];

    __shared__ unsigned taken[NPTS / 32]; // 2 KB bitmask
    __shared__ float sv[8];
    __shared__ int   si[8];
    for (int i = tid; i < NPTS / 32; i += 256) taken[i] = 0u;
    __syncthreads();

    for (int k = 0; k < NEIGH; ++k) {
        float best = 3.4e38f; int bi = 0x40000000;
        for (int j = 0; j < NPTS / 256; ++j) {
            int i = tid + j * 256;
            if (!((taken[i >> 5] >> (i & 31)) & 1u)) {
                float dx = X[i] - ax, dy = Y[i] - ay, dz = Z[i] - az;
                float d = dx * dx + dy * dy + dz * dz;
                if (d < best || (d == best && i < bi)) { best = d; bi = i; }
            }
        }
        argmin_wave(best, bi);            // shuffle, no barrier
        if (lane == 0) { sv[wid] = best; si[wid] = bi; }
        __syncthreads();                  // barrier 1: partials visible
        if (wid == 0) {
            float v  = (lane < 8) ? sv[lane] : 3.4e38f;
            int   i2 = (lane < 8) ? si[lane] : 0x40000000;
            argmin_wave(v, i2);
            if (lane == 0) {
                taken[i2 >> 5] |= 1u << (i2 & 31);
                float wx = X[i2], wy = Y[i2], wz = Z[i2];
                size_t o1 = ((size_t)b * OUT_S + (size_t)t * NEIGH + k) * 3;        // roi1
                size_t o2 = ((size_t)b * OUT_S + 512 + (size_t)t * NEIGH + k) * 3;  // roi2
                out[o1 + 0] = wx; out[o1 + 1] = wy; out[o1 + 2] = wz;
                out[o2 + 0] = wx; out[o2 + 1] = wy; out[o2 + 2] = wz;
            }
        }
        __syncthreads();                  // barrier 2: taken update visible
    }
}

// ---------------------------------------------------------------------------
// Kernel 6: gather the 512 FPS points into the final output slot.
// ---------------------------------------------------------------------------
__global__ void fpsgather_kernel(const float* __restrict__ pts,
                                 const int* __restrict__ fps_idx,
                                 float* __restrict__ out) {
    int g = blockIdx.x * blockDim.x + threadIdx.x;
    if (g >= BATCH * COURSE) return;
    int b = g / COURSE, j = g % COURSE;
    const float* X = pts + (size_t)(b * 3 + 0) * NPTS;
    const float* Y = pts + (size_t)(b * 3 + 1) * NPTS;
    const float* Z = pts + (size_t)(b * 3 + 2) * NPTS;
    int i = fps_idx[g];
    size_t o = ((size_t)b * OUT_S + 1024 + j) * 3;
    out[o + 0] = X[i]; out[o + 1] = Y[i]; out[o + 2] = Z[i];
}

// ---------------------------------------------------------------------------
// Host-side launcher
// ---------------------------------------------------------------------------
extern "C" void kernel_launch(void* const* d_in, const int* in_sizes, int n_in,
                              void* d_out, int out_size, void* d_ws, size_t ws_size,
                              hipStream_t stream) {
    (void)in_sizes; (void)n_in; (void)out_size; (void)ws_size;
    const float* pts = (const float*)d_in[0];     // [16, 3, 16384] f32
    float* out = (float*)d_out;                   // [16, 1536, 3] f32

    // Workspace layout (all 4B-aligned; ~53 KB total)
    int*   fps_idx = (int*)d_ws;                          // BATCH*COURSE
    float* anch    = (float*)(fps_idx + BATCH * COURSE);  // BATCH*APAD*3
    float* anorm   = anch + (size_t)BATCH * APAD * 3;     // BATCH*APAD
    int*   counts  = (int*)(anorm + (size_t)BATCH * APAD);// BATCH*APAD
    int*   sel     = counts + (size_t)BATCH * APAD;       // BATCH*TOPK

    fps_kernel<<<BATCH, 1024, 0, stream>>>(pts, fps_idx);
    anchor_kernel<<<BATCH, APAD, 0, stream>>>(pts, fps_idx, anch, anorm);
    count_kernel<<<dim3(BATCH, 4), 32, 0, stream>>>(pts, anch, anorm, counts);
    top4_kernel<<<1, BATCH, 0, stream>>>(counts, sel);
    knn_kernel<<<dim3(BATCH, TOPK), 256, 0, stream>>>(pts, anch, sel, out);
    fpsgather_kernel<<<(BATCH * COURSE + 255) / 256, 256, 0, stream>>>(pts, fps_idx, out);
}